// GraphLayer_67903432949860
// MI455X (gfx1250) — compile-verified
//
#include <hip/hip_runtime.h>
#include <math.h>

// ---------------------------------------------------------------------------
// GraphLayer: m = H @ W ; agg = scatter_add(m[src] -> dst) ;
//             out = LayerNorm(H + gelu(agg)) * gamma + beta
// B=2, N=50000, D=128, E=800000.  Memory-bound (~2.2 GB traffic -> ~95us @
// 23.3 TB/s).  GEMM (3.3 GFLOP fp32) done with V_WMMA_F32_16X16X4_F32.
// ---------------------------------------------------------------------------

typedef __attribute__((ext_vector_type(2))) float v2f;
typedef __attribute__((ext_vector_type(8))) float v8f;

#define DIM 128
#define LDSS 132   // LDS row stride in floats: 132*4B = 528B = 33*16B ->
                   // float4-aligned rows, and 4-bank rotation per row kills
                   // the 16-way conflict on fragment reads.

// ---------------------------------------------------------------------------
// Kernel 1: M = H * W   (H: [R,128] row-major, W: [128,128] row-major)
// grid.x = R/16 tiles of 16 rows; block = 256 threads = 8 waves.
// Wave w computes output columns [16w, 16w+16) for the block's 16 rows.
// ---------------------------------------------------------------------------
__global__ __launch_bounds__(256)
void gemm_wmma_f32_kernel(const float* __restrict__ H,
                          const float* __restrict__ W,
                          float* __restrict__ M) {
  __shared__ float As[16 * LDSS];

  const int tileM = blockIdx.x;
  const int t     = threadIdx.x;

  // Stage the 16x128 A tile (H rows) into LDS: 512 float4s, 2 per thread.
  const float* Hbase = H + (size_t)tileM * 16 * DIM;
  #pragma unroll
  for (int s = 0; s < 2; ++s) {
    int i   = t + s * 256;          // 0..511
    int row = i >> 5;               // 0..15
    int c4  = i & 31;               // 0..31
    float4 v = reinterpret_cast<const float4*>(Hbase + (size_t)row * DIM)[c4];
    *reinterpret_cast<float4*>(&As[row * LDSS + c4 * 4]) = v;
  }
  __syncthreads();

  const int wave  = t >> 5;
  const int lane  = t & 31;
  const int mrow  = lane & 15;           // A-matrix M index for this lane
  const int khalf = (lane >> 4) << 1;    // lanes 0-15: K+{0,1}; 16-31: K+{2,3}
  const int col   = (wave << 4) + (lane & 15);  // B/C/D N index

  v8f c = {};
  #pragma unroll
  for (int k0 = 0; k0 < DIM; k0 += 4) {
    // A fragment (16x4 fp32): ISA 7.12.2 layout.
    v2f a;
    a.x = As[mrow * LDSS + k0 + khalf + 0];
    a.y = As[mrow * LDSS + k0 + khalf + 1];
    // B fragment (4x16 fp32): row-of-B striped across lanes per VGPR.
    v2f b;
    b.x = W[(size_t)(k0 + khalf + 0) * DIM + col];
    b.y = W[(size_t)(k0 + khalf + 1) * DIM + col];
    // D = A*B + C, fp32 in / fp32 accum. EXEC is all-1s here (no divergence).
    c = __builtin_amdgcn_wmma_f32_16x16x4_f32(
        /*neg_a=*/false, a, /*neg_b=*/false, b,
        /*c_mod=*/(short)0, c, /*reuse_a=*/false, /*reuse_b=*/false);
  }

  // C/D layout: VGPR v -> M = v + 8*(lane>=16); N = lane%16.
  const int rbase = tileM * 16 + ((lane >> 4) << 3);
  #pragma unroll
  for (int v = 0; v < 8; ++v) {
    M[(size_t)(rbase + v) * DIM + col] = c[v];
  }
}

// ---------------------------------------------------------------------------
// Kernel 2: zero agg (float4 stores)
// ---------------------------------------------------------------------------
__global__ __launch_bounds__(256)
void zero_kernel(float4* __restrict__ p, int n4) {
  int i = blockIdx.x * 256 + threadIdx.x;
  if (i < n4) p[i] = make_float4(0.f, 0.f, 0.f, 0.f);
}

// ---------------------------------------------------------------------------
// Kernel 3: agg[b, dst[e], :] += M[b, src[e], :]
// 64 threads per edge: sub>>5 = batch, (sub&31)*4 = feature offset (float4).
// unsafeAtomicAdd -> global_atomic_add_f32 (no CAS loop); resolves in L2.
// ---------------------------------------------------------------------------
__global__ __launch_bounds__(256)
void scatter_kernel(const float* __restrict__ M,
                    const int* __restrict__ src,
                    const int* __restrict__ dst,
                    float* __restrict__ agg,
                    int E, int N) {
  int t = blockIdx.x * 256 + threadIdx.x;
  int e = t >> 6;
  if (e >= E) return;
  int sub = t & 63;
  int b   = sub >> 5;
  int j   = (sub & 31) << 2;

  size_t soff = ((size_t)b * N + (size_t)src[e]) * DIM + j;
  size_t doff = ((size_t)b * N + (size_t)dst[e]) * DIM + j;

  float4 v = *reinterpret_cast<const float4*>(M + soff);
  unsafeAtomicAdd(agg + doff + 0, v.x);
  unsafeAtomicAdd(agg + doff + 1, v.y);
  unsafeAtomicAdd(agg + doff + 2, v.z);
  unsafeAtomicAdd(agg + doff + 3, v.w);
}

// ---------------------------------------------------------------------------
// Kernel 4: out = LayerNorm(H + gelu_exact(agg)) * gamma + beta
// One wave per row (lane holds 4 of the 128 features); butterfly shfl_xor
// reductions across the 32-lane wave.
// ---------------------------------------------------------------------------
__device__ __forceinline__ float gelu_exact(float x) {
  return 0.5f * x * (1.0f + erff(x * 0.70710678118654752f));
}

__global__ __launch_bounds__(256)
void finalize_kernel(const float* __restrict__ H,
                     const float* __restrict__ agg,
                     const float* __restrict__ gamma,
                     const float* __restrict__ beta,
                     float* __restrict__ out,
                     int R) {
  int row  = blockIdx.x * 8 + (threadIdx.x >> 5);
  int lane = threadIdx.x & 31;
  if (row >= R) return;

  size_t base = (size_t)row * DIM + lane * 4;
  float4 h = *reinterpret_cast<const float4*>(H + base);
  float4 a = *reinterpret_cast<const float4*>(agg + base);

  float x0 = h.x + gelu_exact(a.x);
  float x1 = h.y + gelu_exact(a.y);
  float x2 = h.z + gelu_exact(a.z);
  float x3 = h.w + gelu_exact(a.w);

  float s  = x0 + x1 + x2 + x3;
  float s2 = x0 * x0 + x1 * x1 + x2 * x2 + x3 * x3;
  #pragma unroll
  for (int m = 16; m >= 1; m >>= 1) {
    s  += __shfl_xor(s,  m, 32);
    s2 += __shfl_xor(s2, m, 32);
  }
  float mean = s * (1.0f / DIM);
  float var  = s2 * (1.0f / DIM) - mean * mean;
  float rstd = rsqrtf(var + 1e-5f);

  float4 g = *reinterpret_cast<const float4*>(gamma + lane * 4);
  float4 bb = *reinterpret_cast<const float4*>(beta + lane * 4);

  float4 y;
  y.x = (x0 - mean) * rstd * g.x + bb.x;
  y.y = (x1 - mean) * rstd * g.y + bb.y;
  y.z = (x2 - mean) * rstd * g.z + bb.z;
  y.w = (x3 - mean) * rstd * g.w + bb.w;
  *reinterpret_cast<float4*>(out + base) = y;
}

// ---------------------------------------------------------------------------
// Launch
// ---------------------------------------------------------------------------
extern "C" void kernel_launch(void* const* d_in, const int* in_sizes, int n_in,
                              void* d_out, int out_size, void* d_ws, size_t ws_size,
                              hipStream_t stream) {
  const float* H     = (const float*)d_in[0];
  const int*   src   = (const int*)  d_in[1];
  const int*   dst   = (const int*)  d_in[2];
  const float* W     = (const float*)d_in[3];
  const float* gamma = (const float*)d_in[4];
  const float* beta  = (const float*)d_in[5];
  float*       out   = (float*)d_out;

  const int E = in_sizes[1];          // 800000
  const int R = in_sizes[0] / DIM;    // B*N = 100000
  const int B = 2;
  const int N = R / B;                // 50000

  float* m   = (float*)d_ws;                 // [R, 128]
  float* agg = m + (size_t)R * DIM;          // [R, 128]

  // 1) m = H * W  (WMMA fp32)
  gemm_wmma_f32_kernel<<<R / 16, 256, 0, stream>>>(H, W, m);

  // 2) agg = 0
  int n4 = (R * DIM) / 4;
  zero_kernel<<<(n4 + 255) / 256, 256, 0, stream>>>(
      reinterpret_cast<float4*>(agg), n4);

  // 3) scatter-add over edges (64 threads / edge)
  long long sthreads = (long long)E * 64;
  scatter_kernel<<<(int)((sthreads + 255) / 256), 256, 0, stream>>>(
      m, src, dst, agg, E, N);

  // 4) gelu + residual + LayerNorm
  finalize_kernel<<<(R + 7) / 8, 256, 0, stream>>>(H, agg, gamma, beta, out, R);
}